// LayoutLinear_20925080666777
// MI455X (gfx1250) — compile-verified
//
#include <hip/hip_runtime.h>
#include <stdint.h>

#define EPB 256   // edges per chunk
#define CPB 4     // chunks per block (pipelined)
#define F   128   // dense feature width (reference: weight is [K,128])

// ---------------------------------------------------------------------------
// Zero the output (harness poisons d_out; atomics need a clean accumulator).
// float4 => global_store_b128, pure store-bandwidth bound.
// ---------------------------------------------------------------------------
__global__ __launch_bounds__(256) void spmm_zero_kernel(float* __restrict__ out,
                                                        long long n) {
  const float4 z = {0.f, 0.f, 0.f, 0.f};
  long long n4 = n >> 2;
  long long i = (long long)blockIdx.x * blockDim.x + threadIdx.x;
  long long stride = (long long)gridDim.x * blockDim.x;
  float4* out4 = (float4*)out;
  for (long long k = i; k < n4; k += stride) out4[k] = z;
  // scalar tail (out_size % 4), handled by the first few threads
  long long tail = n & 3LL;
  if (i < tail) out[n4 * 4 + i] = 0.f;
}

// ---------------------------------------------------------------------------
// COO SpMM: out[rows[e], :] += values[e] * weight[cols[e], :]
//
// Each block owns CPB consecutive 256-edge chunks and runs a 2-deep async
// pipeline: while waves chew on chunk c (one 512B weight-row gather per edge
// + non-returning global_atomic_add_f32 scatter, all L2-resident on MI455X:
// weight 51.2MB + out 51.2MB << 192MB L2), chunk c+1 streams from HBM into
// the other LDS buffer via global_load_async_to_lds_b128. In-order ASYNCcnt
// decrement lets `s_wait_asynccnt 1` retire chunk c's copies while chunk
// c+1's stay in flight.
// ---------------------------------------------------------------------------
__global__ __launch_bounds__(256) void spmm_coo_kernel(
    const float* __restrict__ values,
    const int*   __restrict__ rows,
    const int*   __restrict__ cols,
    const float* __restrict__ weight,
    float*       __restrict__ out,
    const long long nnz,
    const int chunks_total)
{
  __shared__ __align__(16) float s_val[2][EPB];
  __shared__ __align__(16) int   s_row[2][EPB];
  __shared__ __align__(16) int   s_col[2][EPB];

  const int tid = threadIdx.x;

  const int cbeg = blockIdx.x * CPB;
  if (cbeg >= chunks_total) return;
  const int cend = (cbeg + CPB < chunks_total) ? (cbeg + CPB) : chunks_total;

  // Stage chunk `c` into buffer `p`. Returns true iff the async path was used
  // (full chunk); tail chunks fall back to guarded synchronous loads.
  auto stage = [&](int c, int p) -> bool {
    const long long base = (long long)c * EPB;
    const long long rem  = nnz - base;
    if (rem >= EPB) {
      // 192 lanes x 16B = 3KB; one wave32 instruction moves 512B to LDS.
      if (tid < 192) {
        uint64_t src;
        uint32_t lds;
        if (tid < 64) {
          src = (uint64_t)(uintptr_t)(values + base + (long long)tid * 4);
          lds = (uint32_t)(uintptr_t)&s_val[p][tid * 4];
        } else if (tid < 128) {
          const int j = tid - 64;
          src = (uint64_t)(uintptr_t)(rows + base + (long long)j * 4);
          lds = (uint32_t)(uintptr_t)&s_row[p][j * 4];
        } else {
          const int j = tid - 128;
          src = (uint64_t)(uintptr_t)(cols + base + (long long)j * 4);
          lds = (uint32_t)(uintptr_t)&s_col[p][j * 4];
        }
        asm volatile("global_load_async_to_lds_b128 %0, %1, off"
                     :: "v"(lds), "v"(src)
                     : "memory");
      }
      return true;
    } else {
      const int n = (int)rem;
      for (int i = tid; i < n; i += 256) {
        s_val[p][i] = values[base + i];
        s_row[p][i] = rows[base + i];
        s_col[p][i] = cols[base + i];
      }
      return false;
    }
  };

  const int lane = tid & 31;   // wave32
  const int wave = tid >> 5;   // 0..7
  const int foff = lane * 4;   // 32 lanes x float4 = 128 features

  int p = 0;
  stage(cbeg, 0);

  for (int c = cbeg; c < cend; ++c) {
    // Prefetch next chunk into the other buffer (WAR-safe: previous consumers
    // of that buffer finished at the trailing __syncthreads of iteration c-1).
    bool next_async = false;
    if (c + 1 < cend) next_async = stage(c + 1, p ^ 1);

    // Retire chunk c's async copies; leave chunk c+1's in flight.
    if (next_async) {
      asm volatile("s_wait_asynccnt 0x1" ::: "memory");
    } else {
      asm volatile("s_wait_asynccnt 0x0" ::: "memory");
    }
    __syncthreads();

    // ---- consume chunk c from buffer p ----
    const long long base = (long long)c * EPB;
    const long long rem  = nnz - base;
    const int nedges = (rem >= EPB) ? EPB : (int)rem;

    const int ebeg = wave * 32;
    const int eend = (ebeg + 32 < nedges) ? (ebeg + 32) : nedges;

    #pragma unroll 4
    for (int e = ebeg; e < eend; ++e) {
      const float v = s_val[p][e];               // LDS broadcast reads
      const long long r  = (long long)s_row[p][e];
      const long long cc = (long long)s_col[p][e];

      // One 512B L2 transaction per wave: global_load_b128 per lane.
      const float4 w = *reinterpret_cast<const float4*>(weight + cc * F + foff);

      const uint64_t oaddr = (uint64_t)(uintptr_t)(out + r * F + foff);
      const float ax = v * w.x;
      const float ay = v * w.y;
      const float az = v * w.z;
      const float aw = v * w.w;
      // Guaranteed non-returning hardware float atomics (STOREcnt,
      // fire-and-forget; s_endpgm's implicit wait-idle drains them).
      asm volatile(
          "global_atomic_add_f32 %0, %1, off\n\t"
          "global_atomic_add_f32 %0, %2, off offset:4\n\t"
          "global_atomic_add_f32 %0, %3, off offset:8\n\t"
          "global_atomic_add_f32 %0, %4, off offset:12"
          :: "v"(oaddr), "v"(ax), "v"(ay), "v"(az), "v"(aw)
          : "memory");
    }
    __syncthreads();   // all waves done with buffer p before it is restaged
    p ^= 1;
  }
}

// ---------------------------------------------------------------------------
extern "C" void kernel_launch(void* const* d_in, const int* in_sizes, int n_in,
                              void* d_out, int out_size, void* d_ws, size_t ws_size,
                              hipStream_t stream) {
  const float* values = (const float*)d_in[0];
  const int*   rows   = (const int*)d_in[1];
  const int*   cols   = (const int*)d_in[2];
  const float* weight = (const float*)d_in[3];
  float*       out    = (float*)d_out;

  const long long nnz = (long long)in_sizes[0];

  // Zero the accumulator.
  {
    long long n = (long long)out_size;
    long long n4 = (n + 3) / 4;
    int blocks = (int)((n4 + 255) / 256);
    if (blocks > 65535) blocks = 65535;
    if (blocks < 1) blocks = 1;
    spmm_zero_kernel<<<blocks, 256, 0, stream>>>(out, n);
  }

  // SpMM scatter, CPB chunks per block with a 2-deep async pipeline.
  {
    const int chunks_total = (int)((nnz + EPB - 1) / EPB);
    const int nblocks = (chunks_total + CPB - 1) / CPB;
    spmm_coo_kernel<<<dim3((unsigned)nblocks), 256, 0, stream>>>(
        values, rows, cols, weight, out, nnz, chunks_total);
  }
}